// LSTMMultiNotes_47433618817460
// MI455X (gfx1250) — compile-verified
//
#include <hip/hip_runtime.h>
#include <hip/hip_bf16.h>
#include <math.h>

// Problem constants (from reference): T=8192, IN=109, H=1024, 4H=4096
#define Tn   8192
#define INn  109
#define Hn   1024
#define G4   4096
#define SCAN_WGS 64   // 64 persistent workgroups; each owns 16 hidden units (64 gate rows)

typedef float v2f __attribute__((ext_vector_type(2)));
typedef float v8f __attribute__((ext_vector_type(8)));
typedef unsigned int u32x4 __attribute__((ext_vector_type(4)));
typedef unsigned int u32x8 __attribute__((ext_vector_type(8)));

// ---------------------------------------------------------------------------
// Workspace layout (bytes):
//   xproj : [8192,4096] f32 = 134217728   (reused by both layers)
//   h0s   : [8192,1024] f32 =  33554432
//   h1s   : [8192,1024] f32 =  33554432
//   hcur  : [2,1024]    f32 =      8192   (double-buffered h vector)
//   ctr   : unsigned                 64
// ---------------------------------------------------------------------------
#define OFF_XPROJ 0ull
#define OFF_H0S   134217728ull
#define OFF_H1S   (134217728ull + 33554432ull)
#define OFF_HCUR  (134217728ull + 2ull*33554432ull)
#define OFF_CTR   (OFF_HCUR + 8192ull)

// ---------------------------------------------------------------------------
// init: zero h double-buffer and grid-sync counter (run before each scan)
// ---------------------------------------------------------------------------
__global__ void init_sync(float* __restrict__ hcur, unsigned* __restrict__ counter) {
    int tid = threadIdx.x;
    for (int i = tid; i < 2048; i += 256) hcur[i] = 0.0f;
    if (tid == 0) *counter = 0u;
}

// ---------------------------------------------------------------------------
// x_proj GEMM:  out[8192,4096] = X[8192,K] @ W[4096,K]^T + b0 + b1
// fp32 WMMA 16x16x4. Grid (8,512), 256 thr = 8 waves; wave computes 16x64.
// Steady-state loop is guard-free; one guarded tail step handles K%4 != 0.
// VEC2: rows are 8B-aligned (K even) -> use float2 loads.
// ---------------------------------------------------------------------------
template <bool VEC2>
__global__ __launch_bounds__(256) void gemm_xproj(
    const float* __restrict__ X, const float* __restrict__ W,
    const float* __restrict__ b0, const float* __restrict__ b1,
    float* __restrict__ out, int K)
{
    const int tid  = threadIdx.x;
    const int lane = tid & 31;
    const int wave = tid >> 5;
    const int half = lane >> 4;        // 0: lanes 0-15, 1: lanes 16-31
    const int l15  = lane & 15;
    const int koff = half * 2;         // A/B: upper half-wave holds K=2,3

    const int m0   = blockIdx.y * 16;              // output row tile
    const int n0   = blockIdx.x * 512 + wave * 64; // output col strip per wave
    const int arow = m0 + l15;

    const float* px = X + (size_t)arow * K + koff;
    const float* p0 = W + (size_t)(n0 +  0 + l15) * K + koff;
    const float* p1 = W + (size_t)(n0 + 16 + l15) * K + koff;
    const float* p2 = W + (size_t)(n0 + 32 + l15) * K + koff;
    const float* p3 = W + (size_t)(n0 + 48 + l15) * K + koff;

    v8f acc0 = {0,0,0,0,0,0,0,0};
    v8f acc1 = {0,0,0,0,0,0,0,0};
    v8f acc2 = {0,0,0,0,0,0,0,0};
    v8f acc3 = {0,0,0,0,0,0,0,0};

    const int Kbody = K & ~3;

#define LD2(P)  (VEC2 ? (*(v2f*)(const float2*)(P)) \
                      : (v2f){(P)[0], (P)[1]})
#define WMMA(A, B, ACC) \
    ACC = __builtin_amdgcn_wmma_f32_16x16x4_f32(false, (A), false, (B), (short)0, (ACC), false, false)

#pragma unroll 4
    for (int k = 0; k < Kbody; k += 4) {
        const v2f a  = LD2(px);
        const v2f w0 = LD2(p0);
        const v2f w1 = LD2(p1);
        const v2f w2 = LD2(p2);
        const v2f w3 = LD2(p3);
        WMMA(a, w0, acc0);
        WMMA(a, w1, acc1);
        WMMA(a, w2, acc2);
        WMMA(a, w3, acc3);
        px += 4; p0 += 4; p1 += 4; p2 += 4; p3 += 4;
    }
#undef LD2

    if (Kbody < K) {   // tail (executed once; K=109 -> k=108, koff selects valid lanes)
        const int kx = Kbody + koff;
        const bool e0 = (kx < K), e1 = (kx + 1 < K);
        v2f a, w0, w1, w2, w3;
        a.x  = e0 ? px[0] : 0.0f;  a.y  = e1 ? px[1] : 0.0f;
        w0.x = e0 ? p0[0] : 0.0f;  w0.y = e1 ? p0[1] : 0.0f;
        w1.x = e0 ? p1[0] : 0.0f;  w1.y = e1 ? p1[1] : 0.0f;
        w2.x = e0 ? p2[0] : 0.0f;  w2.y = e1 ? p2[1] : 0.0f;
        w3.x = e0 ? p3[0] : 0.0f;  w3.y = e1 ? p3[1] : 0.0f;
        WMMA(a, w0, acc0);
        WMMA(a, w1, acc1);
        WMMA(a, w2, acc2);
        WMMA(a, w3, acc3);
    }
#undef WMMA

    const int mrow = m0 + half * 8;    // D layout: VGPR v <-> M = v + 8*half
#define STORE_TILE(JJ, ACC)                                                  \
    {                                                                        \
        const int col  = n0 + (JJ)*16 + l15;                                 \
        const float bias = b0[col] + b1[col];                                \
        for (int v = 0; v < 8; ++v)                                          \
            out[(size_t)(mrow + v) * G4 + col] = ACC[v] + bias;              \
    }
    STORE_TILE(0, acc0)
    STORE_TILE(1, acc1)
    STORE_TILE(2, acc2)
    STORE_TILE(3, acc3)
#undef STORE_TILE
}

// ---------------------------------------------------------------------------
// TDM: DMA a 2D tile (tile_dim1=16 rows x tile_dim0=1024 f32) of the
// 4096x1024 W_hh tensor into LDS at lds_off. D# groups 2/3 are NULL (2D).
// Group0: count=1, lds_addr[63:32], global_addr[120:64], type=2[127:126].
// Group1: data_size=4B[17:16], tensor_dim0[79:48], tensor_dim1[111:80],
//         tile_dim0[127:112], tile_dim1[143:128], tensor_dim0_stride[207:160].
// ---------------------------------------------------------------------------
__device__ __forceinline__ void tdm_load_tile_16x1024(unsigned lds_off,
                                                      const float* gaddr) {
    const unsigned long long ga = (unsigned long long)(size_t)gaddr;
    u32x4 g0;
    g0[0] = 1u;                                            // count=1, user mode
    g0[1] = lds_off;                                       // lds_addr (bytes)
    g0[2] = (unsigned)(ga & 0xffffffffu);                  // global_addr[31:0]
    g0[3] = (unsigned)((ga >> 32) & 0x01ffffffu)           // global_addr[56:32]
          | (2u << 30);                                    // type=2 ("image")
    u32x8 g1;
    g1[0] = 2u << 16;                                      // data_size=2 (4 bytes)
    g1[1] = 1024u << 16;                                   // tensor_dim0=1024 (bits 63:48)
    g1[2] = (4096u & 0xffffu) << 16;                       // tensor_dim1 lo16 (bits 95:80)
    g1[3] = 1024u << 16;                                   // tile_dim0=1024 (bits 127:112)
    g1[4] = 16u;                                           // tile_dim1=16 (bits 143:128)
    g1[5] = 1024u;                                         // tensor_dim0_stride lo32
    g1[6] = 0u;
    g1[7] = 0u;
    asm volatile("tensor_load_to_lds %0, %1" :: "s"(g0), "s"(g1) : "memory");
}

// ---------------------------------------------------------------------------
// Persistent LSTM scan. 64 WGs x 256 threads; WG w owns hidden units
// [16w,16w+16) => 64 gate rows of W_hh, TDM-loaded into LDS (256 KB).
// One device-wide split barrier per timestep (arrive-and-spin on an
// agent-scope monotonic counter); h double-buffered in global (L2).
// ---------------------------------------------------------------------------
__global__ __launch_bounds__(256, 1) void lstm_scan(
    const float* __restrict__ Whh,    // [4096,1024]
    const float* __restrict__ xproj,  // [T,4096]  (i,f,g,o blocks of 1024)
    float* __restrict__ hs,           // [T,1024]  per-step h output
    float* __restrict__ hcur,         // [2,1024]  double buffer
    unsigned int* __restrict__ counter,
    int T)
{
    __shared__ float Wl[64][1024];    // 256 KB weight slice
    __shared__ float hbuf[1024];
    __shared__ float gates[64];       // local rows: [0,16)=i [16,32)=f [32,48)=g [48,64)=o
    __shared__ float cbuf[16];

    const int tid  = threadIdx.x;
    const int lane = tid & 31;
    const int wave = tid >> 5;
    const int wg   = blockIdx.x;

    // Preload W_hh slice via Tensor Data Mover: 4 tiles (one per gate block),
    // each 16 rows x 4 KB, gathered from rows g*1024 + wg*16 .. +15.
    if (wave == 0) {
        const unsigned ldsW = (unsigned)(size_t)(void*)&Wl[0][0];
        for (int g = 0; g < 4; ++g)
            tdm_load_tile_16x1024(ldsW + (unsigned)g * 65536u,
                                  Whh + ((size_t)g * 1024 + (size_t)wg * 16) * 1024);
        __builtin_amdgcn_s_wait_tensorcnt(0);
    }
    if (tid < 16) cbuf[tid] = 0.0f;
    __syncthreads();

    for (int t = 0; t < T; ++t) {
        // Load h_t (buffer t&1) with agent-scope loads (L2-coherent), into LDS.
        {
            const float* hsrc = hcur + (t & 1) * 1024;
            float4 h4;
            h4.x = __hip_atomic_load(hsrc + 4 * tid + 0, __ATOMIC_RELAXED, __HIP_MEMORY_SCOPE_AGENT);
            h4.y = __hip_atomic_load(hsrc + 4 * tid + 1, __ATOMIC_RELAXED, __HIP_MEMORY_SCOPE_AGENT);
            h4.z = __hip_atomic_load(hsrc + 4 * tid + 2, __ATOMIC_RELAXED, __HIP_MEMORY_SCOPE_AGENT);
            h4.w = __hip_atomic_load(hsrc + 4 * tid + 3, __ATOMIC_RELAXED, __HIP_MEMORY_SCOPE_AGENT);
            ((float4*)hbuf)[tid] = h4;
        }
        // Prefetch next step's x_proj slice (4 gate segments) -> global_prefetch_b8
        if (tid < 4 && (t + 1) < T)
            __builtin_prefetch(&xproj[(size_t)(t + 1) * G4 + tid * 1024 + wg * 16], 0, 3);
        __syncthreads();

        // GEMV: each wave handles rows {wave, wave+8, ...}; lanes stride the K dim.
        for (int i = 0; i < 8; ++i) {
            const int lr = (i << 3) + wave;
            const float4* wr = (const float4*)&Wl[lr][0];
            const float4* hv = (const float4*)hbuf;
            float s = 0.0f;
#pragma unroll
            for (int j = 0; j < 8; ++j) {
                const float4 w4 = wr[lane + (j << 5)];
                const float4 h4 = hv[lane + (j << 5)];
                s = fmaf(w4.x, h4.x, fmaf(w4.y, h4.y, fmaf(w4.z, h4.z, fmaf(w4.w, h4.w, s))));
            }
#pragma unroll
            for (int off = 16; off; off >>= 1) s += __shfl_xor(s, off, 32);
            if (lane == 0) {
                const int col = (lr >> 4) * 1024 + wg * 16 + (lr & 15);
                gates[lr] = s + xproj[(size_t)t * G4 + col];
            }
        }
        __syncthreads();

        // Pointwise LSTM cell update for this WG's 16 hidden units.
        if (tid < 16) {
            const float iv = 1.0f / (1.0f + expf(-gates[tid]));
            const float fv = 1.0f / (1.0f + expf(-gates[16 + tid]));
            const float gv = tanhf(gates[32 + tid]);
            const float ov = 1.0f / (1.0f + expf(-gates[48 + tid]));
            const float c  = fv * cbuf[tid] + iv * gv;
            cbuf[tid] = c;
            const float hv = ov * tanhf(c);
            const int gu = wg * 16 + tid;
            __hip_atomic_store(&hcur[((t + 1) & 1) * 1024 + gu], hv,
                               __ATOMIC_RELAXED, __HIP_MEMORY_SCOPE_AGENT);
            hs[(size_t)t * Hn + gu] = hv;
        }
        if (wave == 0) __threadfence();   // writers' stores device-visible
        __syncthreads();

        // Device-wide split barrier: arrive, then spin on monotonic counter.
        if (tid == 0) {
            __hip_atomic_fetch_add(counter, 1u, __ATOMIC_RELEASE, __HIP_MEMORY_SCOPE_AGENT);
            const unsigned target = (unsigned)(t + 1) * SCAN_WGS;
            while (__hip_atomic_load(counter, __ATOMIC_ACQUIRE, __HIP_MEMORY_SCOPE_AGENT) < target)
                __builtin_amdgcn_s_sleep(2);
        }
        __syncthreads();
    }
}

// ---------------------------------------------------------------------------
// Final FC + log_softmax: out[t,c] = logsoftmax(h1[t,:] @ fc_w[c,:] + fc_b[c])
// One block (128 thr) per timestep; fc_w (446 KB) stays L2-resident.
// ---------------------------------------------------------------------------
__global__ __launch_bounds__(128) void fc_logsoftmax(
    const float* __restrict__ h1s, const float* __restrict__ fcw,
    const float* __restrict__ fcb, float* __restrict__ out)
{
    __shared__ float hb[1024];
    __shared__ float red[8];

    const int t = blockIdx.x, tid = threadIdx.x;
    const int lane = tid & 31, wid = tid >> 5;

    const float4* hsrc = (const float4*)(h1s + (size_t)t * 1024);
    ((float4*)hb)[tid]       = hsrc[tid];
    ((float4*)hb)[tid + 128] = hsrc[tid + 128];
    __syncthreads();

    float logit = 0.0f;
    float s = -3.402823466e38f;
    if (tid < INn) {
        const float4* wr = (const float4*)(fcw + (size_t)tid * 1024);
        const float4* hv = (const float4*)hb;
        float acc = fcb[tid];
        for (int j = 0; j < 256; ++j) {
            const float4 w4 = wr[j];
            const float4 h4 = hv[j];
            acc = fmaf(w4.x, h4.x, fmaf(w4.y, h4.y, fmaf(w4.z, h4.z, fmaf(w4.w, h4.w, acc))));
        }
        logit = acc;
        s = acc;
    }
    // max over 109
    for (int off = 16; off; off >>= 1) s = fmaxf(s, __shfl_xor(s, off, 32));
    if (lane == 0) red[wid] = s;
    __syncthreads();
    const float m = fmaxf(fmaxf(red[0], red[1]), fmaxf(red[2], red[3]));
    // sum exp
    float e = (tid < INn) ? expf(logit - m) : 0.0f;
    for (int off = 16; off; off >>= 1) e += __shfl_xor(e, off, 32);
    if (lane == 0) red[4 + wid] = e;
    __syncthreads();
    const float lse = logf(red[4] + red[5] + red[6] + red[7]);
    if (tid < INn) out[(size_t)t * INn + tid] = logit - m - lse;
}

// ---------------------------------------------------------------------------
extern "C" void kernel_launch(void* const* d_in, const int* in_sizes, int n_in,
                              void* d_out, int out_size, void* d_ws, size_t ws_size,
                              hipStream_t stream) {
    const float* x    = (const float*)d_in[0];   // [8192,109]
    const float* Wih0 = (const float*)d_in[1];   // [4096,109]
    const float* Whh0 = (const float*)d_in[2];   // [4096,1024]
    const float* bih0 = (const float*)d_in[3];   // [4096]
    const float* bhh0 = (const float*)d_in[4];   // [4096]
    const float* Wih1 = (const float*)d_in[5];   // [4096,1024]
    const float* Whh1 = (const float*)d_in[6];   // [4096,1024]
    const float* bih1 = (const float*)d_in[7];   // [4096]
    const float* bhh1 = (const float*)d_in[8];   // [4096]
    const float* fcw  = (const float*)d_in[9];   // [109,1024]
    const float* fcb  = (const float*)d_in[10];  // [109]
    float* out = (float*)d_out;

    char* ws = (char*)d_ws;
    float*    xproj   = (float*)(ws + OFF_XPROJ);
    float*    h0s     = (float*)(ws + OFF_H0S);
    float*    h1s     = (float*)(ws + OFF_H1S);
    float*    hcur    = (float*)(ws + OFF_HCUR);
    unsigned* counter = (unsigned*)(ws + OFF_CTR);

    const dim3 ggemm(8, 512);

    // Layer 0 (K=109: unaligned rows -> scalar loads)
    init_sync<<<1, 256, 0, stream>>>(hcur, counter);
    gemm_xproj<false><<<ggemm, 256, 0, stream>>>(x, Wih0, bih0, bhh0, xproj, INn);
    lstm_scan<<<SCAN_WGS, 256, 0, stream>>>(Whh0, xproj, h0s, hcur, counter, Tn);

    // Layer 1 (K=1024: 8B-aligned rows -> float2 loads)
    init_sync<<<1, 256, 0, stream>>>(hcur, counter);
    gemm_xproj<true><<<ggemm, 256, 0, stream>>>(h0s, Wih1, bih1, bhh1, xproj, Hn);
    lstm_scan<<<SCAN_WGS, 256, 0, stream>>>(Whh1, xproj, h1s, hcur, counter, Tn);

    // FC + log_softmax
    fc_logsoftmax<<<Tn, 128, 0, stream>>>(h1s, fcw, fcb, out);
}